// DKVMN_63282048139903
// MI455X (gfx1250) — compile-verified
//
#include <hip/hip_runtime.h>
#include <hip/hip_bf16.h>
#include <math.h>

// ---------------------------------------------------------------------------
// DKVMN on MI455X (gfx1250, wave32).
//  Phase 1: state-independent math as f32 WMMA GEMMs (v_wmma_f32_16x16x4_f32):
//           input projections, correlation logits, erase/add gates.
//  Phase 2: 200-step scan, 1 workgroup per batch element, vmat + lin_w in LDS.
//           Only stores h (classifier does NOT feed back into state).
//  Phase 3: classifier out = sigmoid(H @ cls_w^T + b) as one big WMMA GEMM.
// ---------------------------------------------------------------------------

#define Bb     64
#define Ss     200
#define INUM   1000
#define IN_DIM 2000
#define CNUM   50
#define EDIM   128
#define BS     (Bb * Ss)         // 12800 rows
#define CPAD   64                // padded correlation-weight leading dim
#define LSTRIDE (EDIM + 1)       // LDS row stride for transposed lin_w (bank-safe)

typedef float v2f __attribute__((ext_vector_type(2)));
typedef float v8f __attribute__((ext_vector_type(8)));

__device__ __forceinline__ float sigmoidf_(float x) {
    return 1.0f / (1.0f + expf(-x));
}

// ---------------------------------------------------------------------------
// C[M, ldc] = act( X[M,K] @ W[N,K]^T + bias )   via V_WMMA_F32_16X16X4_F32
// One wave computes a 16 x (16*NT) tile; 8 waves/block; blockIdx.y selects the
// n-tile group. B-row pointers are CLAMPED (not zero-guarded) so all loads are
// unconditional -> EXEC stays all-ones.  Explicit DEPTH-2 software pipeline
// with two disjoint register buffers so regalloc cannot serialize the loads
// behind the WMMAs (round-2 disasm showed s_wait_loadcnt 0x0 chains when the
// two pipeline stages shared registers).
// ACT: 0 = none, 1 = sigmoid, 2 = tanh.  Requires K % 8 == 0.
// ---------------------------------------------------------------------------
template <int NT, int ACT>
__global__ __launch_bounds__(256) void wmma_gemm_xt(
    const float* __restrict__ X, const float* __restrict__ W,
    const float* __restrict__ bias, float* __restrict__ C,
    int K, int N, int ldc)
{
    const int wave = threadIdx.x >> 5;             // 0..7
    const int lane = threadIdx.x & 31;
    const int half = lane >> 4;                    // K-split (0/1)
    const int lm   = lane & 15;                    // M (A) or N (B) index
    const int m0   = (blockIdx.x * 8 + wave) * 16;
    const int nt0  = blockIdx.y * NT;              // first n-tile of this wave
    const int ko0  = 2 * half;

    const float* xrow = X + (size_t)(m0 + lm) * (size_t)K;
    const float* wrow[NT];
#pragma unroll
    for (int nt = 0; nt < NT; ++nt) {
        int n = (nt0 + nt) * 16 + lm;
        if (n >= N) n = N - 1;                     // clamp: loads stay unguarded
        wrow[nt] = W + (size_t)n * (size_t)K;
    }

    v8f acc[NT];
#pragma unroll
    for (int nt = 0; nt < NT; ++nt) acc[nt] = (v8f){};

    // --- depth-2 pipeline prologue: stage0 (k=0), stage1 (k=4) -------------
    v2f a0 = *(const v2f*)(xrow + ko0);
    v2f a1 = *(const v2f*)(xrow + 4 + ko0);
    v2f b0[NT], b1[NT];
#pragma unroll
    for (int nt = 0; nt < NT; ++nt) {
        b0[nt] = *(const v2f*)(wrow[nt] + ko0);
        b1[nt] = *(const v2f*)(wrow[nt] + 4 + ko0);
    }

    // --- steady state: consume stage k0 / k0+4, prefetch k0+8 / k0+12 -----
    for (int k0 = 0; k0 < K - 8; k0 += 8) {
        {   // stage A
            const v2f au = a0;
            a0 = *(const v2f*)(xrow + k0 + 8 + ko0);
#pragma unroll
            for (int nt = 0; nt < NT; ++nt) {
                acc[nt] = __builtin_amdgcn_wmma_f32_16x16x4_f32(
                    false, au, false, b0[nt], (short)0, acc[nt], false, false);
                b0[nt] = *(const v2f*)(wrow[nt] + k0 + 8 + ko0);
            }
        }
        {   // stage B
            const v2f au = a1;
            a1 = *(const v2f*)(xrow + k0 + 12 + ko0);
#pragma unroll
            for (int nt = 0; nt < NT; ++nt) {
                acc[nt] = __builtin_amdgcn_wmma_f32_16x16x4_f32(
                    false, au, false, b1[nt], (short)0, acc[nt], false, false);
                b1[nt] = *(const v2f*)(wrow[nt] + k0 + 12 + ko0);
            }
        }
    }
    // --- epilogue: last two stages, no prefetch ----------------------------
#pragma unroll
    for (int nt = 0; nt < NT; ++nt)
        acc[nt] = __builtin_amdgcn_wmma_f32_16x16x4_f32(
            false, a0, false, b0[nt], (short)0, acc[nt], false, false);
#pragma unroll
    for (int nt = 0; nt < NT; ++nt)
        acc[nt] = __builtin_amdgcn_wmma_f32_16x16x4_f32(
            false, a1, false, b1[nt], (short)0, acc[nt], false, false);

    // --- store: bias + activation (skip pad columns) -----------------------
#pragma unroll
    for (int nt = 0; nt < NT; ++nt) {
        const int n = (nt0 + nt) * 16 + lm;
        if (n < N) {
            const float bv = (bias != nullptr) ? bias[n] : 0.0f;
#pragma unroll
            for (int v = 0; v < 8; ++v) {
                float x = acc[nt][v] + bv;
                if (ACT == 1) x = sigmoidf_(x);
                if (ACT == 2) x = tanhf(x);
                C[(size_t)(m0 + v + 8 * half) * (size_t)ldc + n] = x;
            }
        }
    }
}

// ---------------------------------------------------------------------------
// In-place row softmax over first CNUM columns of a [BS, CPAD] buffer.
// ---------------------------------------------------------------------------
__global__ __launch_bounds__(256) void softmax_rows(float* __restrict__ Wm)
{
    const int row = blockIdx.x * blockDim.x + threadIdx.x;
    if (row >= BS) return;
    float* p = Wm + (size_t)row * CPAD;
    float mx = -INFINITY;
    for (int c = 0; c < CNUM; ++c) mx = fmaxf(mx, p[c]);
    float sum = 0.0f;
    for (int c = 0; c < CNUM; ++c) sum += expf(p[c] - mx);
    const float inv = 1.0f / sum;
    for (int c = 0; c < CNUM; ++c) p[c] = expf(p[c] - mx) * inv;
}

// ---------------------------------------------------------------------------
// Sequential scan: one block (8 wave32) per batch element.
//  - vmat (25.6 KB) and lin_w (transposed, padded; 132 KB) live in LDS.
//  - per step: r = w.vmat ; h = tanh(lin([r;itm])) -> stored to H ;
//              vmat = vmat*(1 - w⊗e) + w⊗a.   (classifier deferred)
// ---------------------------------------------------------------------------
__global__ __launch_bounds__(256) void dkvmn_scan(
    const float* __restrict__ ITM,   // [BS, EDIM]
    const float* __restrict__ Wsm,   // [BS, CPAD] softmaxed correlation weights
    const float* __restrict__ EM,    // [BS, EDIM] erase gates (sigmoid)
    const float* __restrict__ AM,    // [BS, EDIM] add vectors (tanh)
    const float* __restrict__ vmat0, // [CNUM, EDIM]
    const float* __restrict__ lin_w, // [EDIM, 2*EDIM]
    const float* __restrict__ lin_b, // [EDIM]
    float* __restrict__ H)           // [BS, EDIM]
{
    __shared__ float vmat[CNUM * EDIM];
    __shared__ float linT[2 * EDIM * LSTRIDE];   // [k][j] stride 129: bank-safe
    __shared__ float sh_w[CPAD];
    __shared__ float sh_itm[EDIM];
    __shared__ float sh_e[EDIM];
    __shared__ float sh_a[EDIM];
    __shared__ float sh_r[EDIM];

    const int b   = blockIdx.x;
    const int tid = threadIdx.x;

    for (int i = tid; i < CNUM * EDIM; i += 256) vmat[i] = vmat0[i];
    for (int idx = tid; idx < 2 * EDIM * EDIM; idx += 256) {
        const int j = idx >> 8;           // output row of lin_w
        const int k = idx & 255;          // input column
        linT[k * LSTRIDE + j] = lin_w[idx];
    }
    const float my_lin_b = (tid < EDIM) ? lin_b[tid] : 0.0f;
    __syncthreads();

    for (int s = 0; s < Ss; ++s) {
        const size_t row = (size_t)b * Ss + s;
        if (tid < EDIM) {
            sh_itm[tid] = ITM[row * EDIM + tid];
            sh_e[tid]   = EM[row * EDIM + tid];
            sh_a[tid]   = AM[row * EDIM + tid];
        } else if (tid < EDIM + CPAD) {
            sh_w[tid - EDIM] = Wsm[row * CPAD + (tid - EDIM)];
        }
        __syncthreads();

        // r = w . vmat   (consecutive tid -> consecutive banks)
        if (tid < EDIM) {
            float acc = 0.0f;
#pragma unroll 5
            for (int c = 0; c < CNUM; ++c) acc += sh_w[c] * vmat[c * EDIM + tid];
            sh_r[tid] = acc;
        }
        __syncthreads();

        // h = tanh(lin_w @ [r ; itm] + lin_b) -> H   (classifier deferred)
        if (tid < EDIM) {
            float acc = my_lin_b;
#pragma unroll 8
            for (int k = 0; k < EDIM; ++k)
                acc += linT[k * LSTRIDE + tid] * sh_r[k];
#pragma unroll 8
            for (int k = 0; k < EDIM; ++k)
                acc += linT[(EDIM + k) * LSTRIDE + tid] * sh_itm[k];
            H[row * EDIM + tid] = tanhf(acc);
        }

        // vmat = vmat*(1 - w⊗e) + w⊗a
        for (int idx = tid; idx < CNUM * EDIM; idx += 256) {
            const int c = idx >> 7;
            const int e = idx & (EDIM - 1);
            const float wv = sh_w[c];
            vmat[idx] = vmat[idx] * (1.0f - wv * sh_e[e]) + wv * sh_a[e];
        }
        __syncthreads();
    }
}

// ---------------------------------------------------------------------------
extern "C" void kernel_launch(void* const* d_in, const int* in_sizes, int n_in,
                              void* d_out, int out_size, void* d_ws, size_t ws_size,
                              hipStream_t stream)
{
    const float* item        = (const float*)d_in[0];   // [B,S,INUM]
    const float* interaction = (const float*)d_in[1];   // [B,S,IN_DIM]
    const float* A_w   = (const float*)d_in[2];         // [EDIM, INUM]
    const float* B_w   = (const float*)d_in[3];         // [EDIM, IN_DIM]
    const float* kmat  = (const float*)d_in[4];         // [CNUM, EDIM]
    const float* vmat0 = (const float*)d_in[5];         // [CNUM, EDIM]
    const float* lin_w = (const float*)d_in[6];         // [EDIM, 2*EDIM]
    const float* lin_b = (const float*)d_in[7];
    const float* cls_w = (const float*)d_in[8];         // [INUM, EDIM]
    const float* cls_b = (const float*)d_in[9];
    const float* er_w  = (const float*)d_in[10];        // [EDIM, EDIM]
    const float* er_b  = (const float*)d_in[11];
    const float* ad_w  = (const float*)d_in[12];        // [EDIM, EDIM]
    const float* ad_b  = (const float*)d_in[13];
    float* out = (float*)d_out;

    float* ws = (float*)d_ws;
    float* ITM  = ws;                              // [BS, EDIM]
    float* ITR  = ITM  + (size_t)BS * EDIM;        // [BS, EDIM] (reused as H)
    float* WLOG = ITR  + (size_t)BS * EDIM;        // [BS, CPAD]
    float* EM   = WLOG + (size_t)BS * CPAD;        // [BS, EDIM]
    float* AM   = EM   + (size_t)BS * EDIM;        // [BS, EDIM]
    float* Hbuf = ITR;                             // ITR dead after EM/AM GEMMs

    const int gemm_blocks = (BS / 16) / 8;         // 100

    // Input projections (big K)
    wmma_gemm_xt<8, 0><<<dim3(gemm_blocks, 1), 256, 0, stream>>>(item,        A_w, nullptr, ITM, INUM,   EDIM, EDIM);
    wmma_gemm_xt<8, 0><<<dim3(gemm_blocks, 1), 256, 0, stream>>>(interaction, B_w, nullptr, ITR, IN_DIM, EDIM, EDIM);

    // State-independent scan pieces hoisted out as GEMMs
    wmma_gemm_xt<4, 0><<<dim3(gemm_blocks, 1), 256, 0, stream>>>(ITM, kmat, nullptr, WLOG, EDIM, CNUM, CPAD);
    wmma_gemm_xt<8, 1><<<dim3(gemm_blocks, 1), 256, 0, stream>>>(ITR, er_w, er_b,    EM,   EDIM, EDIM, EDIM);
    wmma_gemm_xt<8, 2><<<dim3(gemm_blocks, 1), 256, 0, stream>>>(ITR, ad_w, ad_b,    AM,   EDIM, EDIM, EDIM);

    softmax_rows<<<BS / 256, 256, 0, stream>>>(WLOG);

    // Sequential part (produces H; classifier deferred out of the scan)
    dkvmn_scan<<<Bb, 256, 0, stream>>>(ITM, WLOG, EM, AM, vmat0,
                                       lin_w, lin_b, Hbuf);

    // Classifier as one big WMMA GEMM: out = sigmoid(H @ cls_w^T + cls_b)
    // N = 1000 -> 63 n-tiles = 9 groups x NT=7
    wmma_gemm_xt<7, 1><<<dim3(gemm_blocks, 9), 256, 0, stream>>>(Hbuf, cls_w, cls_b, out, EDIM, INUM, INUM);
}